// RNN_45303315038370
// MI455X (gfx1250) — compile-verified
//
#include <hip/hip_runtime.h>
#include <hip/hip_bf16.h>

// Elman RNN (B=262144, T=10, I=1, H=32) on gfx1250.
// Per wave32: 16 batch rows. Hnew^T = Whh x Hprev^T via chained
// V_WMMA_F32_16X16X4_F32 (8 K-steps x 2 output halves = 16 WMMA/step),
// full fp32 precision, recurrence kept entirely in VGPRs.

typedef __attribute__((ext_vector_type(2))) float v2f;
typedef __attribute__((ext_vector_type(8))) float v8f;

#define T_STEPS 10
#define HID     32

__device__ __forceinline__ float fast_tanh(float v) {
#if __has_builtin(__builtin_amdgcn_tanhf)
    return __builtin_amdgcn_tanhf(v);
#elif __has_builtin(__builtin_amdgcn_tanh_f32)
    return __builtin_amdgcn_tanh_f32(v);
#else
    return tanhf(v);
#endif
}

// Fetch Hnew[k][b] out of the D-layout tanh results (lo: n=0..15, hi: n=16..31).
// D layout: reg r, lanes 0-15 -> n=r, lanes 16-31 -> n=r+8 (per 16-row tile).
__device__ __forceinline__ float fetch_h(const float lo[8], const float hi[8],
                                         int k, int b) {
    float s = (k < 16) ? lo[k & 7] : hi[k & 7];
    int srcLane = b + (((k & 15) >= 8) ? 16 : 0);
    return __shfl(s, srcLane, 32);
}

__global__ __launch_bounds__(256) void rnn_wmma_f32_kernel(
    const float* __restrict__ x,     // [B,T] (I==1)
    const float* __restrict__ h0,    // [B,H]
    const float* __restrict__ Wih,   // [H] (I==1)
    const float* __restrict__ Whh,   // [H,H] row-major
    const float* __restrict__ bih,   // [H]
    const float* __restrict__ bhh,   // [H]
    const float* __restrict__ Wout,  // [H]
    const float* __restrict__ bout,  // [1]
    float* __restrict__ outs,        // [B,T]
    float* __restrict__ hlast)       // [B,H]
{
    const int lane = threadIdx.x & 31;
    const int wave = threadIdx.x >> 5;
    const int tile = blockIdx.x * 8 + wave;   // 16 batch rows per wave
    const int base = tile * 16;
    const int b    = lane & 15;               // batch-in-tile (lanes) / A row m
    const int hf   = lane >> 4;               // lane half
    const int row  = base + b;

    // ---- per-lane constants (D layout: reg r <-> n = r + 8*hf [+16]) ----
    float wA[8], wB[8], bA[8], bB[8], oA[8], oB[8];
#pragma unroll
    for (int r = 0; r < 8; ++r) {
        const int n = r + 8 * hf;
        wA[r] = Wih[n];            wB[r] = Wih[n + 16];
        bA[r] = bih[n] + bhh[n];   bB[r] = bih[n + 16] + bhh[n + 16];
        oA[r] = Wout[n];           oB[r] = Wout[n + 16];
    }
    const float bo = bout[0];

    // ---- Whh as A fragments: A'[m][k] = Whh[m][k]; 8 K=4 chunks x 2 halves
    // A 16x4 f32 layout: v0 = {K=4c (lanes<16), K=4c+2 (lanes>=16)}, v1 = +1.
    v2f aLo[8], aHi[8];
#pragma unroll
    for (int c = 0; c < 8; ++c) {
        const int k = 4 * c + 2 * hf;
        aLo[c].x = Whh[b * HID + k];
        aLo[c].y = Whh[b * HID + k + 1];
        aHi[c].x = Whh[(b + 16) * HID + k];
        aHi[c].y = Whh[(b + 16) * HID + k + 1];
    }

    // ---- B fragments = Hprev^T: B'[k][b]=H[b][k]; init from h0 ----
    v2f bf[8];
#pragma unroll
    for (int c = 0; c < 8; ++c) {
        const int k = 4 * c + 2 * hf;
        bf[c].x = h0[row * HID + k];
        bf[c].y = h0[row * HID + k + 1];
    }

    // ---- preload x for this row ----
    float xr[T_STEPS];
#pragma unroll
    for (int t = 0; t < T_STEPS; ++t) xr[t] = x[row * T_STEPS + t];

    float hlo[8], hhi[8];

#pragma unroll
    for (int t = 0; t < T_STEPS; ++t) {
        // Fold input projection + biases into the WMMA accumulator init.
        v8f dlo, dhi;
#pragma unroll
        for (int r = 0; r < 8; ++r) {
            dlo[r] = fmaf(xr[t], wA[r], bA[r]);
            dhi[r] = fmaf(xr[t], wB[r], bB[r]);
        }

        // Hnew^T[0:16][:]  += Whh[0:16][:]  x Hprev^T  (8 chained K=4 steps)
#pragma unroll
        for (int c = 0; c < 8; ++c)
            dlo = __builtin_amdgcn_wmma_f32_16x16x4_f32(
                false, aLo[c], false, bf[c], (short)0, dlo, false, false);
        // Hnew^T[16:32][:] += Whh[16:32][:] x Hprev^T
#pragma unroll
        for (int c = 0; c < 8; ++c)
            dhi = __builtin_amdgcn_wmma_f32_16x16x4_f32(
                false, aHi[c], false, bf[c], (short)0, dhi, false, false);

        // tanh + fused output head (per-lane partial dot over n)
        float p = 0.0f;
#pragma unroll
        for (int r = 0; r < 8; ++r) {
            hlo[r] = fast_tanh(dlo[r]);
            hhi[r] = fast_tanh(dhi[r]);
            p = fmaf(hlo[r], oA[r], p);
            p = fmaf(hhi[r], oB[r], p);
        }
        p += __shfl_xor(p, 16, 32);            // combine the two lane halves
        if (lane < 16) outs[row * T_STEPS + t] = p + bo;

        // Re-layout D -> B' fragments for the next timestep (cross-lane).
        if (t + 1 < T_STEPS) {
#pragma unroll
            for (int c = 0; c < 8; ++c) {
                const float f0 = fetch_h(hlo, hhi, 4 * c + 0, b);
                const float f1 = fetch_h(hlo, hhi, 4 * c + 1, b);
                const float f2 = fetch_h(hlo, hhi, 4 * c + 2, b);
                const float f3 = fetch_h(hlo, hhi, 4 * c + 3, b);
                bf[c].x = hf ? f2 : f0;
                bf[c].y = hf ? f3 : f1;
            }
        }
    }

    // ---- h_last: lane (b,hf) covers n = r+8*hf and n+16 ----
#pragma unroll
    for (int r = 0; r < 8; ++r) {
        const int n = r + 8 * hf;
        hlast[row * HID + n]      = hlo[r];
        hlast[row * HID + n + 16] = hhi[r];
    }
}

extern "C" void kernel_launch(void* const* d_in, const int* in_sizes, int n_in,
                              void* d_out, int out_size, void* d_ws, size_t ws_size,
                              hipStream_t stream) {
    const float* x    = (const float*)d_in[0];
    const float* h0   = (const float*)d_in[1];
    const float* Wih  = (const float*)d_in[2];
    const float* Whh  = (const float*)d_in[3];
    const float* bih  = (const float*)d_in[4];
    const float* bhh  = (const float*)d_in[5];
    const float* Wout = (const float*)d_in[6];
    const float* bout = (const float*)d_in[7];

    const int Bn = in_sizes[0] / T_STEPS;       // 262144
    float* outs  = (float*)d_out;               // [B,T,1] flat
    float* hlast = (float*)d_out + (size_t)Bn * T_STEPS;  // [1,B,H] flat

    const int blocks = Bn / 128;                // 8 waves x 16 rows per block
    rnn_wmma_f32_kernel<<<blocks, 256, 0, stream>>>(
        x, h0, Wih, Whh, bih, bhh, Wout, bout, outs, hlast);
}